// IndustryGNN_90263032692924
// MI455X (gfx1250) — compile-verified
//
#include <hip/hip_runtime.h>
#include <cstdint>

#define HID 128

typedef __attribute__((ext_vector_type(2))) float v2f;
typedef __attribute__((ext_vector_type(8))) float v8f;

// ---------------------------------------------------------------- utilities
__global__ __launch_bounds__(256) void zero_f4(float4* __restrict__ p, long n4) {
    long i = (long)blockIdx.x * blockDim.x + threadIdx.x;
    if (i < n4) p[i] = make_float4(0.f, 0.f, 0.f, 0.f);
}

__global__ __launch_bounds__(256) void degree_kernel(const int* __restrict__ dst,
                                                     float* __restrict__ deg,
                                                     int E, int N) {
    int i = blockIdx.x * blockDim.x + threadIdx.x;
    if (i >= E) return;
    int d = dst[i];
    d = min(max(d, 0), N - 1);
    unsafeAtomicAdd(&deg[d], 1.0f);          // native global_atomic_add_f32
}

__global__ __launch_bounds__(256) void rsqrt_kernel(const float* __restrict__ deg,
                                                    float* __restrict__ dis, int N) {
    int i = blockIdx.x * blockDim.x + threadIdx.x;
    if (i < N) dis[i] = rsqrtf(deg[i] + 1.0f);   // +1 = self loop
}

// ------------------------------------------------- dense transform: T = H @ W
// One block = 16 rows; 8 waves each own a 16-wide column tile of the 128 cols.
// K=128 swept with 32 x V_WMMA_F32_16X16X4_F32.
__global__ __launch_bounds__(256) void gemm_nodes(const float* __restrict__ Hin,
                                                  const float* __restrict__ W,
                                                  float* __restrict__ T) {
    __shared__ float sH[16][HID + 2];          // pad -> no LDS bank conflicts
    const int tid  = threadIdx.x;
    const int wave = tid >> 5;
    const int lane = tid & 31;
    const int row0 = blockIdx.x * 16;

    for (int i = tid; i < 16 * HID; i += 256) {
        int r = i >> 7, k = i & (HID - 1);
        sH[r][k] = Hin[(size_t)(row0 + r) * HID + k];
    }
    __syncthreads();

    const int col0 = wave * 16;
    const int half = lane >> 4;                // 0: K+0/+1, 1: K+2/+3
    const int lr   = lane & 15;

    v8f c = {};
    for (int k0 = 0; k0 < HID; k0 += 4) {
        const int ka = k0 + 2 * half;
        v2f a, b;
        a.x = sH[lr][ka];
        a.y = sH[lr][ka + 1];
        b.x = W[(size_t)ka * HID + col0 + lr];
        b.y = W[(size_t)(ka + 1) * HID + col0 + lr];
        c = __builtin_amdgcn_wmma_f32_16x16x4_f32(false, a, false, b,
                                                  (short)0, c, false, false);
    }
#pragma unroll
    for (int r = 0; r < 8; ++r)
        T[(size_t)(row0 + r + 8 * half) * HID + col0 + lr] = c[r];
}

// ------------------------------------------- edge scatter: A[dst] += T[src]*n
// One wave per edge: 32 lanes x float4 = one 128-wide feature row.
__global__ __launch_bounds__(256) void scatter_edges(const int* __restrict__ src,
                                                     const int* __restrict__ dst,
                                                     const float* __restrict__ dis,
                                                     const float* __restrict__ T,
                                                     float* __restrict__ A,
                                                     int E, int N) {
    int wave = blockIdx.x * (blockDim.x >> 5) + (threadIdx.x >> 5);
    if (wave >= E) return;
    int lane = threadIdx.x & 31;
    int s = src[wave], d = dst[wave];
    s = min(max(s, 0), N - 1);
    d = min(max(d, 0), N - 1);
    float nrm = dis[s] * dis[d];
    float4 v = ((const float4*)(T + (size_t)s * HID))[lane];
    float* a = A + (size_t)d * HID + lane * 4;
    unsafeAtomicAdd(a + 0, v.x * nrm);       // global_atomic_add_f32, no CAS loop
    unsafeAtomicAdd(a + 1, v.y * nrm);
    unsafeAtomicAdd(a + 2, v.z * nrm);
    unsafeAtomicAdd(a + 3, v.w * nrm);
}

// ---------------- fused self-loop + bias + activation, in place: A = act(...)
__global__ __launch_bounds__(256) void finish_kernel(float* __restrict__ A,
                                                     const float* __restrict__ T,
                                                     const float* __restrict__ dis,
                                                     const float* __restrict__ bias,
                                                     long n, int relu) {
    long i = (long)blockIdx.x * blockDim.x + threadIdx.x;
    if (i >= n) return;
    int node = (int)(i >> 7);
    int f    = (int)(i & (HID - 1));
    float dd = dis[node];
    float v  = A[i] + T[i] * dd * dd + bias[f];
    A[i] = relu ? fmaxf(v, 0.0f) : v;
}

// ------------------------------------- classifier head: Out = H @ Wlin + blin
// 4 waves per block, each wave one 16-row tile; Ncls = 16 exactly one WMMA tile.
__global__ __launch_bounds__(128) void gemm_out(const float* __restrict__ Hin,
                                                const float* __restrict__ Wlin,
                                                const float* __restrict__ blin,
                                                float* __restrict__ Out,
                                                int ntiles) {
    __shared__ float sH[4][16][HID + 2];
    const int tid  = threadIdx.x;
    const int wave = tid >> 5;
    const int lane = tid & 31;
    const int tile = blockIdx.x * 4 + wave;
    const bool active = tile < ntiles;
    const int row0 = tile * 16;

    if (active) {
        for (int i = lane; i < 16 * HID; i += 32) {
            int r = i >> 7, k = i & (HID - 1);
            sH[wave][r][k] = Hin[(size_t)(row0 + r) * HID + k];
        }
    }
    __syncthreads();
    if (!active) return;

    const int half = lane >> 4;
    const int lr   = lane & 15;
    v8f c = {};
    for (int k0 = 0; k0 < HID; k0 += 4) {
        const int ka = k0 + 2 * half;
        v2f a, b;
        a.x = sH[wave][lr][ka];
        a.y = sH[wave][lr][ka + 1];
        b.x = Wlin[(size_t)ka * 16 + lr];
        b.y = Wlin[(size_t)(ka + 1) * 16 + lr];
        c = __builtin_amdgcn_wmma_f32_16x16x4_f32(false, a, false, b,
                                                  (short)0, c, false, false);
    }
    float bb = blin[lr];
#pragma unroll
    for (int r = 0; r < 8; ++r)
        Out[(size_t)(row0 + r + 8 * half) * 16 + lr] = c[r] + bb;
}

// ---------------------------------------------------------------- host side
extern "C" void kernel_launch(void* const* d_in, const int* in_sizes, int n_in,
                              void* d_out, int out_size, void* d_ws, size_t ws_size,
                              hipStream_t stream) {
    const float* x  = (const float*)d_in[0];
    const int*   ei = (const int*)  d_in[1];
    const float* Wk[3] = { (const float*)d_in[2], (const float*)d_in[4], (const float*)d_in[6] };
    const float* bk[3] = { (const float*)d_in[3], (const float*)d_in[5], (const float*)d_in[7] };
    const float* Wl = (const float*)d_in[8];
    const float* bl = (const float*)d_in[9];

    const int N = in_sizes[0] / HID;      // 50000
    const int E = in_sizes[1] / 2;        // 600000
    const int* srcIdx = ei;               // edge_index[0]
    const int* dstIdx = ei + E;           // edge_index[1]

    // workspace carve-out (256B aligned): deg, dis, T, A   (~52 MB)
    char* w = (char*)d_ws;
    auto take = [&](size_t bytes) {
        char* p = w; w += (bytes + 255) & ~(size_t)255; return p;
    };
    float* deg = (float*)take((size_t)N * sizeof(float));
    float* dis = (float*)take((size_t)N * sizeof(float));
    float* T   = (float*)take((size_t)N * HID * sizeof(float));
    float* A   = (float*)take((size_t)N * HID * sizeof(float));

    const long nh     = (long)N * HID;
    const int  ntiles = N / 16;                       // 3125 (N % 16 == 0)

    // --- normalization: deg -> dis = rsqrt(deg + 1) ---
    zero_f4<<<(N / 4 + 255) / 256, 256, 0, stream>>>((float4*)deg, N / 4);
    degree_kernel<<<(E + 255) / 256, 256, 0, stream>>>(dstIdx, deg, E, N);
    rsqrt_kernel<<<(N + 255) / 256, 256, 0, stream>>>(deg, dis, N);

    // --- three GCN layers ---
    for (int layer = 0; layer < 3; ++layer) {
        const float* Hin = (layer == 0) ? x : A;
        gemm_nodes<<<ntiles, 256, 0, stream>>>(Hin, Wk[layer], T);
        zero_f4<<<(int)((nh / 4 + 255) / 256), 256, 0, stream>>>((float4*)A, nh / 4);
        scatter_edges<<<(E + 7) / 8, 256, 0, stream>>>(srcIdx, dstIdx, dis, T, A, E, N);
        finish_kernel<<<(int)((nh + 255) / 256), 256, 0, stream>>>(
            A, T, dis, bk[layer], nh, layer < 2 ? 1 : 0);
    }

    // --- classifier head ---
    gemm_out<<<(ntiles + 3) / 4, 128, 0, stream>>>(A, Wl, bl, (float*)d_out, ntiles);
}